// IPUNeighborListMD_23330262352092
// MI455X (gfx1250) — compile-verified
//
#include <hip/hip_runtime.h>

// CDNA5 / gfx1250, wave32.
typedef float v8f __attribute__((ext_vector_type(8)));
typedef float v2f __attribute__((ext_vector_type(2)));

#define MOLS   64
#define NATOMS 512
#define TILE   16
#define RC2    49.0f   // (cutoff + shell)^2 = 7^2

// One block per (molecule, 16-row i-tile): grid = 64*32 = 2048 blocks, 256 thr.
// d2(i,j) = |ri|^2 + |rj|^2 - 2*G(i,j); G = P P^T per 16x16 tile via
// V_WMMA_F32_16X16X4_F32 (K=3 padded to 4 with zeros).
//
// Note: no explicit i != j exclusion is needed. On the diagonal Rij = 0
// exactly, so the masked output is 0 regardless of the mask bit.
__global__ __launch_bounds__(256) void ipu_nbl_kernel(
    const float* __restrict__ pos,   // [64,512,3]
    float* __restrict__ out)         // [64,512,512,3]
{
    __shared__ float sx[NATOMS], sy[NATOMS], sz[NATOMS], sn[NATOMS];

    const int m   = (int)(blockIdx.x >> 5);        // molecule
    const int i0  = (int)(blockIdx.x & 31) * TILE; // i-tile base
    const int tid = (int)threadIdx.x;

    // Stage this molecule's positions + squared norms in LDS (8 KB).
    const float* pm = pos + (size_t)m * NATOMS * 3;
    for (int a = tid; a < NATOMS; a += 256) {
        const float x = pm[a * 3 + 0];
        const float y = pm[a * 3 + 1];
        const float z = pm[a * 3 + 2];
        sx[a] = x; sy[a] = y; sz[a] = z;
        sn[a] = x * x + y * y + z * z;
    }
    __syncthreads();

    const int  lane   = tid & 31;   // wave32 lane id
    // Wave id is wave-invariant: pin it to an SGPR so all loop/address math
    // stays scalar (no exec-mask control flow around the WMMAs).
    const int  wave   = __builtin_amdgcn_readfirstlane(tid >> 5);
    const int  l      = lane & 15;
    const bool hihalf = (lane >= 16);

    // A operand (16x4 f32): rows i0..i0+15, K = {x, y, z, 0}.
    //   lanes 0-15 : v0 = x[row], v1 = y[row];  lanes 16-31: v0 = z[row], v1 = 0
    v2f A;
    A.x = hihalf ? sz[i0 + l] : sx[i0 + l];
    A.y = hihalf ? 0.0f       : sy[i0 + l];

    // Hoist this lane's 8 i-rows (D layout: VGPR v -> row ib+v) into registers.
    const int ib = i0 + (hihalf ? 8 : 0);
    float ni[8], rix[8], riy[8], riz[8];
#pragma unroll
    for (int v = 0; v < 8; ++v) {
        ni[v]  = sn[ib + v];
        rix[v] = sx[ib + v];
        riy[v] = sy[ib + v];
        riz[v] = sz[ib + v];
    }

    const size_t rowstride = (size_t)NATOMS * 3;
    float* const out0 = out + ((size_t)m * NATOMS + ib) * rowstride;

    // Each wave covers j-tiles {wave, wave+8, wave+16, wave+24}; fully
    // unrolled straight-line code: 4 WMMAs + 32 b96 stores per wave.
#pragma unroll
    for (int t = 0; t < 4; ++t) {
        const int j0 = (wave + t * 8) * TILE;

        // B operand (4x16 f32): columns j0..j0+15, same K packing as A.
        v2f B;
        B.x = hihalf ? sz[j0 + l] : sx[j0 + l];
        B.y = hihalf ? 0.0f       : sy[j0 + l];

        v8f C = {};
        v8f G = __builtin_amdgcn_wmma_f32_16x16x4_f32(
            /*neg_a=*/false, A, /*neg_b=*/false, B,
            /*c_mod=*/(short)0, C, /*reuse_a=*/false, /*reuse_b=*/false);

        const int   j   = j0 + l;
        const float nj  = sn[j];
        const float pjx = sx[j], pjy = sy[j], pjz = sz[j];

#pragma unroll
        for (int v = 0; v < 8; ++v) {
            const float d2 = ni[v] + nj - 2.0f * G[v];
            const bool  ok = (d2 < RC2);
            float* p = out0 + (size_t)v * rowstride + 3 * (size_t)j;
            // 16 lanes of a half-wave cover 192 contiguous bytes per row.
            p[0] = ok ? (pjx - rix[v]) : 0.0f;
            p[1] = ok ? (pjy - riy[v]) : 0.0f;
            p[2] = ok ? (pjz - riz[v]) : 0.0f;
        }
    }
}

extern "C" void kernel_launch(void* const* d_in, const int* in_sizes, int n_in,
                              void* d_out, int out_size, void* d_ws, size_t ws_size,
                              hipStream_t stream) {
    (void)in_sizes; (void)n_in; (void)d_ws; (void)ws_size; (void)out_size;
    const float* pos = (const float*)d_in[0];
    float*       out = (float*)d_out;
    const dim3 grid(MOLS * (NATOMS / TILE));  // 2048 blocks
    const dim3 block(256);                    // 8 wave32 waves
    ipu_nbl_kernel<<<grid, block, 0, stream>>>(pos, out);
}